// GCNModelVAE_53644141527286
// MI455X (gfx1250) — compile-verified
//
#include <hip/hip_runtime.h>
#include <hip/hip_bf16.h>

#define GN   8192
#define GE   262144
#define GF   512
#define GH1  32
#define GH2  16

typedef __attribute__((ext_vector_type(2))) float v2f;
typedef __attribute__((ext_vector_type(8))) float v8f;

// D = A(16x4,f32) * B(4x16,f32) + C(16x16,f32)   — V_WMMA_F32_16X16X4_F32
__device__ __forceinline__ v8f wmma4f32(v2f a, v2f b, v8f c) {
    return __builtin_amdgcn_wmma_f32_16x16x4_f32(
        /*neg_a=*/false, a, /*neg_b=*/false, b,
        /*c_mod=*/(short)0, c, /*reuse_a=*/false, /*reuse_b=*/false);
}

// ---------------- zero scratch accumulators ----------------
__global__ void k_zero(float* __restrict__ p, int n) {
    int i = blockIdx.x * blockDim.x + threadIdx.x;
    if (i < n) p[i] = 0.0f;
}

// ---------------- XW = X[8192x512] @ Wh[512x32] (WMMA f32) ----------------
// one wave per 16x16 output tile; tiles: 512(m) x 2(n) = 1024; 8 waves/block
__global__ void k_gemm_xw(const float* __restrict__ X, const float* __restrict__ Wh,
                          float* __restrict__ XW) {
    const int lane = threadIdx.x & 31;
    const int wave = threadIdx.x >> 5;
    const int tile = blockIdx.x * 8 + wave;            // 0..1023
    const int m0 = (tile >> 1) << 4;
    const int n0 = (tile & 1) << 4;
    const int l  = lane & 15;
    const int hi = lane >> 4;                          // 0 or 1

    v8f c = {};
    const float* arow = X + (size_t)(m0 + l) * GF + 2 * hi;   // A[l][kk+2hi .. +1]
    const float* bcol = Wh + n0 + l;
    for (int kk = 0; kk < GF; kk += 4) {
        v2f a = *(const v2f*)(arow + kk);
        v2f b;
        b.x = bcol[(kk + 2 * hi) * GH1];               // B row kk+2hi
        b.y = bcol[(kk + 2 * hi + 1) * GH1];           // B row kk+2hi+1
        c = wmma4f32(a, b, c);
    }
    const int base = (m0 + 8 * hi) * GH1 + n0 + l;
#pragma unroll
    for (int r = 0; r < 8; ++r) XW[base + r * GH1] = c[r];
}

// ---------------- h[row] += w * XW[col]  (32 wide, fp32 HW atomics) ----------------
__global__ void k_spmm_h(const int* __restrict__ row, const int* __restrict__ col,
                         const float* __restrict__ w, const float* __restrict__ XW,
                         float* __restrict__ h) {
    int e = blockIdx.x * blockDim.x + threadIdx.x;
    if (e >= GE) return;
    const int r = row[e], cc = col[e];
    const float wt = w[e];
    const float* src = XW + (size_t)cc * GH1;
    float* dst = h + (size_t)r * GH1;
#pragma unroll
    for (int j = 0; j < GH1; ++j) unsafeAtomicAdd(dst + j, wt * src[j]);
}

// ---------------- hm/hl = relu(h) @ Wmean/Wlogstd  (K=32, tiny) ----------------
__global__ void k_hgemm(const float* __restrict__ h, const float* __restrict__ Wm,
                        const float* __restrict__ Wl, float* __restrict__ hm,
                        float* __restrict__ hl) {
    int idx = blockIdx.x * blockDim.x + threadIdx.x;   // row*16 + j
    int r = idx >> 4, j = idx & 15;
    const float* hr = h + (size_t)r * GH1;
    float sm = 0.f, sl = 0.f;
#pragma unroll
    for (int k = 0; k < GH1; ++k) {
        float v = fmaxf(hr[k], 0.0f);
        sm = fmaf(v, Wm[k * GH2 + j], sm);
        sl = fmaf(v, Wl[k * GH2 + j], sl);
    }
    hm[idx] = sm; hl[idx] = sl;
}

// ---------------- zm/zl[row] += w * hm/hl[col]  (16 wide) ----------------
__global__ void k_spmm_z(const int* __restrict__ row, const int* __restrict__ col,
                         const float* __restrict__ w, const float* __restrict__ hm,
                         const float* __restrict__ hl, float* __restrict__ zm,
                         float* __restrict__ zl) {
    int e = blockIdx.x * blockDim.x + threadIdx.x;
    if (e >= GE) return;
    const int r = row[e], cc = col[e];
    const float wt = w[e];
    const float* sm = hm + (size_t)cc * GH2;
    const float* sl = hl + (size_t)cc * GH2;
    float* dm = zm + (size_t)r * GH2;
    float* dl = zl + (size_t)r * GH2;
#pragma unroll
    for (int j = 0; j < GH2; ++j) unsafeAtomicAdd(dm + j, wt * sm[j]);
#pragma unroll
    for (int j = 0; j < GH2; ++j) unsafeAtomicAdd(dl + j, wt * sl[j]);
}

// ---------------- z = zm + eps * exp(zl) ----------------
__global__ void k_reparam(const float* __restrict__ zm, const float* __restrict__ zl,
                          const float* __restrict__ eps, float* __restrict__ z) {
    int i = blockIdx.x * blockDim.x + threadIdx.x;
    z[i] = fmaf(eps[i], __expf(zl[i]), zm[i]);
}

// ---------------- output = zm1 @ W_out + b_out ----------------
__global__ void k_outhead(const float* __restrict__ zm1, const float* __restrict__ Wo,
                          const float* __restrict__ bo, float* __restrict__ out) {
    int idx = blockIdx.x * blockDim.x + threadIdx.x;   // row*16 + j
    int r = idx >> 4, j = idx & 15;
    float s = bo[j];
#pragma unroll
    for (int k = 0; k < GH2; ++k) s = fmaf(zm1[r * GH2 + k], Wo[k * GH2 + j], s);
    out[idx] = s;
}

// ---------------- recon = z @ z^T  (WMMA f32, K=16, NT streaming stores) ----------
// grid: (64, 512); wave (8/block) -> n-tile = blockIdx.x*8+wave, m-tile = blockIdx.y
__global__ void k_recon(const float* __restrict__ z, float* __restrict__ out) {
    const int lane = threadIdx.x & 31;
    const int wave = threadIdx.x >> 5;
    const int l  = lane & 15;
    const int hi = lane >> 4;
    const int m0 = blockIdx.y << 4;
    const int n0 = ((blockIdx.x << 3) + wave) << 4;

    v8f c = {};
    const float* arow = z + (size_t)(m0 + l) * GH2 + 2 * hi;  // A[l][kk+2hi], +1
    const float* brow = z + (size_t)(n0 + l) * GH2 + 2 * hi;  // B[kk+2hi][l] = z[n0+l][..]
#pragma unroll
    for (int kk = 0; kk < GH2; kk += 4) {
        v2f a = *(const v2f*)(arow + kk);
        v2f b = *(const v2f*)(brow + kk);
        c = wmma4f32(a, b, c);
    }
    size_t base = (size_t)(m0 + 8 * hi) * GN + (size_t)(n0 + l);
#pragma unroll
    for (int r = 0; r < 8; ++r)
        __builtin_nontemporal_store(c[r], out + base + (size_t)r * GN);
}

extern "C" void kernel_launch(void* const* d_in, const int* in_sizes, int n_in,
                              void* d_out, int out_size, void* d_ws, size_t ws_size,
                              hipStream_t stream) {
    const float* X1   = (const float*)d_in[0];
    const int*   row1 = (const int*)  d_in[1];
    const int*   col1 = (const int*)  d_in[2];
    const float* w1   = (const float*)d_in[3];
    const float* X2   = (const float*)d_in[4];
    const int*   row2 = (const int*)  d_in[5];
    const int*   col2 = (const int*)  d_in[6];
    const float* w2   = (const float*)d_in[7];
    const float* eps1 = (const float*)d_in[8];
    const float* eps2 = (const float*)d_in[9];
    const float* Wh1  = (const float*)d_in[10];
    const float* Wm1  = (const float*)d_in[11];
    const float* Wl1  = (const float*)d_in[12];
    const float* Wh2  = (const float*)d_in[13];
    const float* Wm2  = (const float*)d_in[14];
    const float* Wl2  = (const float*)d_in[15];
    const float* Wo   = (const float*)d_in[16];
    const float* bo   = (const float*)d_in[17];

    float* out = (float*)d_out;

    // workspace layout (floats)
    float* ws   = (float*)d_ws;
    float* XW1  = ws;                       // 262144
    float* XW2  = XW1  + GN * GH1;          // 262144
    float* h1   = XW2  + GN * GH1;          // 262144  (zeroed)
    float* h2   = h1   + GN * GH1;          // 262144  (zeroed)
    float* zm1  = h2   + GN * GH1;          // 131072  (zeroed)
    float* zl1  = zm1  + GN * GH2;          // 131072  (zeroed)
    float* zm2  = zl1  + GN * GH2;          // 131072  (zeroed)
    float* zl2  = zm2  + GN * GH2;          // 131072  (zeroed)
    float* hm1  = zl2  + GN * GH2;
    float* hl1  = hm1  + GN * GH2;
    float* hm2  = hl1  + GN * GH2;
    float* hl2  = hm2  + GN * GH2;
    float* zz1  = hl2  + GN * GH2;
    float* zz2  = zz1  + GN * GH2;

    const int B = 256;

    // 1) zero the atomic accumulators (h1,h2,zm1,zl1,zm2,zl2 are contiguous)
    const int nzero = 2 * GN * GH1 + 4 * GN * GH2;     // 1,048,576
    k_zero<<<(nzero + B - 1) / B, B, 0, stream>>>(h1, nzero);

    // 2) dense fp32 WMMA GEMMs: XW = X @ Wh   (1024 tiles -> 128 blocks)
    k_gemm_xw<<<128, B, 0, stream>>>(X1, Wh1, XW1);
    k_gemm_xw<<<128, B, 0, stream>>>(X2, Wh2, XW2);

    // 3) edge scatter: h += w * XW[col]
    k_spmm_h<<<GE / B, B, 0, stream>>>(row1, col1, w1, XW1, h1);
    k_spmm_h<<<GE / B, B, 0, stream>>>(row2, col2, w2, XW2, h2);

    // 4) relu + small GEMMs
    k_hgemm<<<GN * GH2 / B, B, 0, stream>>>(h1, Wm1, Wl1, hm1, hl1);
    k_hgemm<<<GN * GH2 / B, B, 0, stream>>>(h2, Wm2, Wl2, hm2, hl2);

    // 5) edge scatter: zm/zl += w * hm/hl[col]
    k_spmm_z<<<GE / B, B, 0, stream>>>(row1, col1, w1, hm1, hl1, zm1, zl1);
    k_spmm_z<<<GE / B, B, 0, stream>>>(row2, col2, w2, hm2, hl2, zm2, zl2);

    // 6) reparameterize
    k_reparam<<<GN * GH2 / B, B, 0, stream>>>(zm1, zl1, eps1, zz1);
    k_reparam<<<GN * GH2 / B, B, 0, stream>>>(zm2, zl2, eps2, zz2);

    // 7) output head: zm1 @ W_out + b_out  -> d_out[2*N*N ..]
    k_outhead<<<GN * GH2 / B, B, 0, stream>>>(zm1, Wo, bo,
                                              out + 2ull * GN * GN);

    // 8) recon = z @ z^T  (the 537 MB streaming part; WMMA + NT stores)
    dim3 rgrid(GN / 16 / 8, GN / 16);                  // (64, 512)
    k_recon<<<rgrid, B, 0, stream>>>(zz1, out);
    k_recon<<<rgrid, B, 0, stream>>>(zz2, out + 1ull * GN * GN);
}